// SAGE_47210280518211
// MI455X (gfx1250) — compile-verified
//
#include <hip/hip_runtime.h>
#include <math.h>

// ---------------------------------------------------------------------------
// GraphSAGE 2-layer + MLP head forward, f32, CDNA5 (gfx1250, wave32).
// Dense layers: V_WMMA_F32_16X16X4_F32, A tiles staged to LDS with
// GLOBAL_LOAD_ASYNC_TO_LDS_B128 (ASYNCcnt), software-pipelined K loop.
// Aggregation (the memory-bound 90%) uses coalesced loads + f32 atomics.
// ---------------------------------------------------------------------------

typedef float v2f __attribute__((ext_vector_type(2)));
typedef float v8f __attribute__((ext_vector_type(8)));

// ---------------------------------------------------------------------------
// Edge scatter: one block per edge, one thread per feature.
// ---------------------------------------------------------------------------
__global__ void scatter_add_kernel(const float* __restrict__ feat,
                                   const int* __restrict__ ei,   // [2, E]
                                   int E, int F,
                                   float* __restrict__ sum,
                                   float* __restrict__ cnt) {
    const int e   = blockIdx.x;
    const int src = ei[e];
    const int tgt = ei[E + e];
    const int c   = threadIdx.x;
    if (c < F) {
        atomicAdd(&sum[(size_t)tgt * F + c], feat[(size_t)src * F + c]);
    }
    if (c == 0) {
        atomicAdd(&cnt[tgt], 1.0f);
    }
}

// In-place mean: sum[i] /= max(cnt[row], 1)
__global__ void mean_kernel(float* __restrict__ sum,
                            const float* __restrict__ cnt,
                            int n, int F) {
    size_t i = (size_t)blockIdx.x * blockDim.x + threadIdx.x;
    size_t total = (size_t)n * F;
    if (i < total) {
        float c = cnt[i / F];
        sum[i] = sum[i] / fmaxf(c, 1.0f);
    }
}

// ---------------------------------------------------------------------------
// out[M,NOUT] = maybe_relu( A·Wl^T + (Aself·Wr^T) + bias )
// Compile-time K / NOUT / flags -> branch-free, fully unrolled, pipelined.
// Block: (32, NOUT/16); wave y owns 16-col tile, 16-row A tile shared in LDS.
// A staging via global_load_async_to_lds_b128 unless RELU_IN (needs VALU).
// Requires M%16==0, NOUT%16==0, K%4==0 (hold for all call sites).
// ---------------------------------------------------------------------------
template <int K, int NOUT, bool HAS_SELF, bool RELU_OUT, bool RELU_IN>
__global__ __launch_bounds__(32 * (NOUT / 16))
void wmma_linear_kernel(const float* __restrict__ A,
                        const float* __restrict__ Aself,
                        const float* __restrict__ Wl,
                        const float* __restrict__ Wr,
                        const float* __restrict__ bias,
                        float* __restrict__ out) {
    constexpr int KP = K + 4;             // pad: rows start 4 banks apart
    constexpr int NW = NOUT / 16;         // waves per block
    __shared__ float ldsA[16 * KP];
    __shared__ float ldsS[HAS_SELF ? 16 * KP : 4];

    const int row0     = blockIdx.x * 16;
    const int tid      = threadIdx.y * 32 + threadIdx.x;
    constexpr int NTHR = 32 * NW;

    if constexpr (!RELU_IN) {
        // --- CDNA5 async DMA staging: 16-byte chunks, KP*4 % 16 == 0 ---
        constexpr int CPR = K / 4;        // float4 chunks per row
        const unsigned baseA = (unsigned)(uintptr_t)&ldsA[0];
        const unsigned baseS = (unsigned)(uintptr_t)&ldsS[0];
        for (int idx = tid; idx < 16 * CPR; idx += NTHR) {
            const int r = idx / CPR;
            const int c = (idx - r * CPR) * 4;
            const unsigned la = baseA + (unsigned)((r * KP + c) * 4);
            const unsigned long long ga =
                (unsigned long long)(uintptr_t)(A + (size_t)(row0 + r) * K + c);
            asm volatile("global_load_async_to_lds_b128 %0, %1, off"
                         :: "v"(la), "v"(ga) : "memory");
            if constexpr (HAS_SELF) {
                const unsigned ls = baseS + (unsigned)((r * KP + c) * 4);
                const unsigned long long gs =
                    (unsigned long long)(uintptr_t)(Aself + (size_t)(row0 + r) * K + c);
                asm volatile("global_load_async_to_lds_b128 %0, %1, off"
                             :: "v"(ls), "v"(gs) : "memory");
            }
        }
        asm volatile("s_wait_asynccnt 0x0" ::: "memory");
    } else {
        // VALU staging (applies ReLU while copying)
        for (int idx = tid; idx < 16 * K; idx += NTHR) {
            const int r = idx / K;
            const int c = idx - r * K;
            ldsA[r * KP + c] = fmaxf(A[(size_t)(row0 + r) * K + c], 0.0f);
        }
    }
    __syncthreads();

    const int lane = threadIdx.x;
    const int half = lane >> 4;                 // K-half select
    const int r16  = lane & 15;
    const int ncol = threadIdx.y * 16 + r16;    // output column for this lane

    const float* __restrict__ wlrow = Wl + (size_t)ncol * K;
    const float* __restrict__ wrrow = HAS_SELF ? (Wr + (size_t)ncol * K) : Wl;

    const int kofs = half * 2;                  // lanes 16-31 take K+2,K+3

    v8f acc = {};

    // software pipeline: fragments for step k+4 load while step k multiplies
    v2f a0 = *(const v2f*)&ldsA[r16 * KP + kofs];
    v2f b0 = *(const v2f*)&wlrow[kofs];
    v2f s0 = {}, t0 = {};
    if constexpr (HAS_SELF) {
        s0 = *(const v2f*)&ldsS[r16 * KP + kofs];
        t0 = *(const v2f*)&wrrow[kofs];
    }

#pragma unroll
    for (int k = 0; k < K; k += 4) {
        v2f a1 = {}, b1 = {}, s1 = {}, t1 = {};
        if (k + 4 < K) {
            const int kk = k + 4 + kofs;
            a1 = *(const v2f*)&ldsA[r16 * KP + kk];
            b1 = *(const v2f*)&wlrow[kk];
            if constexpr (HAS_SELF) {
                s1 = *(const v2f*)&ldsS[r16 * KP + kk];
                t1 = *(const v2f*)&wrrow[kk];
            }
        }
        acc = __builtin_amdgcn_wmma_f32_16x16x4_f32(false, a0, false, b0,
                                                    (short)0, acc, false, false);
        if constexpr (HAS_SELF) {
            acc = __builtin_amdgcn_wmma_f32_16x16x4_f32(false, s0, false, t0,
                                                        (short)0, acc, false, false);
        }
        a0 = a1; b0 = b1; s0 = s1; t0 = t1;
    }

    // C/D layout: VGPR v -> M = v + 8*half, N = ncol
    const float bv = bias[ncol];
#pragma unroll
    for (int v = 0; v < 8; ++v) {
        const int m = v + half * 8;
        float val = acc[v] + bv;
        if constexpr (RELU_OUT) val = fmaxf(val, 0.0f);
        out[(size_t)(row0 + m) * NOUT + ncol] = val;
    }
}

// ---------------------------------------------------------------------------
// log_softmax over rows of 128. One wave32 per row, 4 cols/lane.
// ---------------------------------------------------------------------------
__global__ void log_softmax_kernel(const float* __restrict__ logits,
                                   float* __restrict__ out, int rows) {
    const int row = blockIdx.x * blockDim.y + threadIdx.y;
    if (row >= rows) return;
    const int lane = threadIdx.x;
    const float* p = logits + (size_t)row * 128;

    float x0 = p[lane], x1 = p[lane + 32], x2 = p[lane + 64], x3 = p[lane + 96];
    float m = fmaxf(fmaxf(x0, x1), fmaxf(x2, x3));
    for (int off = 16; off > 0; off >>= 1) m = fmaxf(m, __shfl_xor(m, off, 32));
    float s = expf(x0 - m) + expf(x1 - m) + expf(x2 - m) + expf(x3 - m);
    for (int off = 16; off > 0; off >>= 1) s += __shfl_xor(s, off, 32);
    const float lse = m + logf(s);

    float* o = out + (size_t)row * 128;
    o[lane]      = x0 - lse;
    o[lane + 32] = x1 - lse;
    o[lane + 64] = x2 - lse;
    o[lane + 96] = x3 - lse;
}

// ---------------------------------------------------------------------------
extern "C" void kernel_launch(void* const* d_in, const int* in_sizes, int n_in,
                              void* d_out, int out_size, void* d_ws, size_t ws_size,
                              hipStream_t stream) {
    const float* x   = (const float*)d_in[0];
    const int*   ei1 = (const int*)  d_in[1];
    const int*   ei2 = (const int*)  d_in[2];
    const float* Wl1 = (const float*)d_in[3];
    const float* bl1 = (const float*)d_in[4];
    const float* Wr1 = (const float*)d_in[5];
    const float* Wl2 = (const float*)d_in[6];
    const float* bl2 = (const float*)d_in[7];
    const float* Wr2 = (const float*)d_in[8];
    const float* W1  = (const float*)d_in[9];
    const float* b1  = (const float*)d_in[10];
    const float* W2  = (const float*)d_in[11];
    const float* b2  = (const float*)d_in[12];

    constexpr int NT1 = 50000, NT2 = 10000;
    constexpr int E1 = 1000000, E2 = 300000;
    constexpr int FIN = 256, FH = 192, FO = 128;

    // workspace layout (floats): ~27.5M floats (~110 MB)
    float* ws = (float*)d_ws;
    size_t o = 0;
    float* agg1   = ws + o; o += (size_t)NT1 * FIN;
    float* cnt1   = ws + o; o += NT1;
    float* h1     = ws + o; o += (size_t)NT1 * FH;
    float* agg2   = ws + o; o += (size_t)NT2 * FH;
    float* cnt2   = ws + o; o += NT2;
    float* h2     = ws + o; o += (size_t)NT2 * FH;
    float* logits = ws + o; o += (size_t)NT2 * FO;

    float* out_ls  = (float*)d_out;                 // [10000,128] log-softmax
    float* out_emb = out_ls + (size_t)NT2 * FO;     // [10000,192] embedding

    hipMemsetAsync(agg1, 0, (size_t)NT1 * FIN * sizeof(float), stream);
    hipMemsetAsync(cnt1, 0, (size_t)NT1 * sizeof(float), stream);
    hipMemsetAsync(agg2, 0, (size_t)NT2 * FH * sizeof(float), stream);
    hipMemsetAsync(cnt2, 0, (size_t)NT2 * sizeof(float), stream);

    // ---- Layer 1: aggregate + SAGE linear (ReLU) ----
    scatter_add_kernel<<<dim3(E1), dim3(FIN), 0, stream>>>(x, ei1, E1, FIN, agg1, cnt1);
    {
        size_t tot = (size_t)NT1 * FIN;
        mean_kernel<<<dim3((unsigned)((tot + 255) / 256)), dim3(256), 0, stream>>>(agg1, cnt1, NT1, FIN);
    }
    wmma_linear_kernel<FIN, FH, true, true, false>
        <<<dim3(NT1 / 16), dim3(32, FH / 16), 0, stream>>>(agg1, x, Wl1, Wr1, bl1, h1);

    // ---- Layer 2: aggregate + SAGE linear (ReLU) ----
    scatter_add_kernel<<<dim3(E2), dim3(FH), 0, stream>>>(h1, ei2, E2, FH, agg2, cnt2);
    {
        size_t tot = (size_t)NT2 * FH;
        mean_kernel<<<dim3((unsigned)((tot + 255) / 256)), dim3(256), 0, stream>>>(agg2, cnt2, NT2, FH);
    }
    wmma_linear_kernel<FH, FH, true, true, false>
        <<<dim3(NT2 / 16), dim3(32, FH / 16), 0, stream>>>(agg2, h1, Wl2, Wr2, bl2, h2);

    // ---- MLP head ----
    // embedding = h2 @ W1^T + b1   (no relu; also an output)
    wmma_linear_kernel<FH, FH, false, false, false>
        <<<dim3(NT2 / 16), dim3(32, FH / 16), 0, stream>>>(h2, nullptr, W1, nullptr, b1, out_emb);
    // logits = relu(embedding) @ W2^T + b2
    wmma_linear_kernel<FH, FO, false, false, true>
        <<<dim3(NT2 / 16), dim3(32, FO / 16), 0, stream>>>(out_emb, nullptr, W2, nullptr, b2, logits);

    // ---- log_softmax rows of 128 ----
    log_softmax_kernel<<<dim3((NT2 + 7) / 8), dim3(32, 8), 0, stream>>>(logits, out_ls, NT2);
}